// NPLinear_25443386261608
// MI455X (gfx1250) — compile-verified
//
#include <hip/hip_runtime.h>

typedef float v2f __attribute__((ext_vector_type(2)));
typedef float v8f __attribute__((ext_vector_type(8)));

// ---------------------------------------------------------------------------
// K0: zero the atomic-accumulator region of the workspace (rm sums + rc sums)
// ---------------------------------------------------------------------------
__global__ void zero_kernel(float* __restrict__ p, long n) {
    long i = (long)blockIdx.x * 256 + threadIdx.x;
    if (i < n) p[i] = 0.f;
}

// ---------------------------------------------------------------------------
// K1: fused reduction over one [O,N] matrix per (b,c):
//   cm[bc][m]     = mean_n w[m,n]        (direct store, one wave owns a row)
//   rm_sum[bc][n] += sum_m w[m,n]        (LDS partials -> global f32 atomics)
//   rc_sum[bc]    += sum_{m,n} w[m,n]    (global f32 atomic)
// ---------------------------------------------------------------------------
template <int O, int N>
__global__ void __launch_bounds__(256) reduce_kernel(const float* __restrict__ w,
                                                     float* __restrict__ rm_sum,
                                                     float* __restrict__ cm,
                                                     float* __restrict__ rc_sum) {
    constexpr int NCH = (N + 31) / 32;
    __shared__ float colsum[N];
    const int bc   = blockIdx.y;
    const int row0 = blockIdx.x * 64;
    const int lane = threadIdx.x & 31;
    const int wave = threadIdx.x >> 5;
    const float* wm = w + (long)bc * O * N;

    for (int n = threadIdx.x; n < N; n += 256) colsum[n] = 0.f;
    __syncthreads();

    float cacc[NCH];
#pragma unroll
    for (int j = 0; j < NCH; ++j) cacc[j] = 0.f;
    float rcloc = 0.f;

    const int rend = (row0 + 64 < O) ? row0 + 64 : O;
    for (int r = row0 + wave; r < rend; r += 8) {
        const float* rp = wm + (long)r * N;
        float rs = 0.f;
#pragma unroll
        for (int j = 0; j < NCH; ++j) {
            int n  = j * 32 + lane;
            float v = (n < N) ? rp[n] : 0.f;
            cacc[j] += v;
            rs += v;
        }
        for (int off = 16; off; off >>= 1) rs += __shfl_xor(rs, off, 32);
        if (lane == 0) {
            cm[(long)bc * O + r] = rs * (1.f / N);
            rcloc += rs;
        }
    }
#pragma unroll
    for (int j = 0; j < NCH; ++j) {
        int n = j * 32 + lane;
        if (n < N) atomicAdd(&colsum[n], cacc[j]);   // ds_add_f32
    }
    if (lane == 0) atomicAdd(&rc_sum[bc], rcloc);    // global_atomic_add_f32
    __syncthreads();
    for (int n = threadIdx.x; n < N; n += 256)
        atomicAdd(&rm_sum[(long)bc * N + n], colsum[n]);
}

// ---------------------------------------------------------------------------
// K2a: wb_means[b][0..95] = concat(rowcol_means x3, bias_means x3)
// ---------------------------------------------------------------------------
__global__ void wb_kernel(const float* __restrict__ rcsum,
                          const float* __restrict__ b0,
                          const float* __restrict__ b1,
                          const float* __restrict__ b2,
                          float* __restrict__ wb) {
    int t = blockIdx.x * 256 + threadIdx.x;
    if (t >= 8 * 96) return;
    int b = t / 96, j = t % 96;
    float v;
    if (j < 48) {
        int layer = j / 16, c = j % 16;
        const float inv[3] = {1.f / (512.f * 784.f), 1.f / (512.f * 512.f), 1.f / (10.f * 512.f)};
        v = rcsum[layer * 128 + b * 16 + c] * inv[layer];
    } else {
        int jj = j - 48;
        int layer = jj / 16, c = jj % 16;
        const float* bp = (layer == 0) ? b0 : (layer == 1) ? b1 : b2;
        int O = (layer == 2) ? 10 : 512;
        float s = 0.f;
        for (int m = 0; m < O; ++m) s += bp[(long)(b * 16 + c) * O + m];
        v = s / O;
    }
    wb[b * 96 + j] = v;
}

// ---------------------------------------------------------------------------
// K2b: z2[i][b][o], u2[i][b][o]: 96-dim dots of wb_means with rc / bias_rc.
// ---------------------------------------------------------------------------
__global__ void z2u2_kernel(const float* __restrict__ wb,
                            const float* rcw0, const float* rcb0,
                            const float* rcw1, const float* rcb1,
                            const float* rcw2, const float* rcb2,
                            const float* brw0, const float* brb0,
                            const float* brw1, const float* brb1,
                            const float* brw2, const float* brb2,
                            float* __restrict__ z2, float* __restrict__ u2) {
    int t = blockIdx.x * 256 + threadIdx.x;
    if (t >= 384) return;
    int layer = t / 128, r = t % 128, b = r / 16, o = r % 16;
    const float* rw = (layer == 0) ? rcw0 : (layer == 1) ? rcw1 : rcw2;
    const float* rb = (layer == 0) ? rcb0 : (layer == 1) ? rcb1 : rcb2;
    const float* bw = (layer == 0) ? brw0 : (layer == 1) ? brw1 : brw2;
    const float* bb = (layer == 0) ? brb0 : (layer == 1) ? brb1 : brb2;
    float s = rb[o], su = bb[o];
    for (int k = 0; k < 96; ++k) {
        float x = wb[b * 96 + k];
        s  += rw[o * 96 + k] * x;
        su += bw[o * 96 + k] * x;
    }
    z2[t] = s;
    u2[t] = su;
}

// ---------------------------------------------------------------------------
// K2c: z3tot[b][o][n] = conv1d_1(row_cat) + r_b[o] + z1_b[o] + z2[b][o]
// ---------------------------------------------------------------------------
__global__ void __launch_bounds__(256) z3_kernel(const float* __restrict__ rm,
                                                 const float* __restrict__ cmprev,
                                                 const float* __restrict__ bprev,
                                                 const float* __restrict__ rw,
                                                 const float* __restrict__ rb,
                                                 const float* __restrict__ z1b,
                                                 const float* __restrict__ z2,
                                                 float* __restrict__ z3out,
                                                 int N, int Oi, int Oprev, int K) {
    int n = blockIdx.x * 256 + threadIdx.x;
    int b = blockIdx.y;
    if (n >= N) return;
    float acc[16];
    for (int o = 0; o < 16; ++o) acc[o] = rb[o] + z1b[o] + z2[b * 16 + o];
    float invO = 1.f / (float)Oi;
    for (int k = 0; k < K; ++k) {
        float x;
        if (k < 16)      x = rm[(long)(b * 16 + k) * N + n] * invO;
        else if (k < 32) x = cmprev[(long)(b * 16 + k - 16) * Oprev + n];
        else             x = bprev[(long)(b * 16 + k - 32) * Oprev + n];
        for (int o = 0; o < 16; ++o) acc[o] += rw[o * K + k] * x;
    }
    for (int o = 0; o < 16; ++o) z3out[(long)(b * 16 + o) * N + n] = acc[o];
}

// ---------------------------------------------------------------------------
// K2d: z4[b][o][m] (to ws) and out_b[b][o][m] = u1 + u2 (to d_out)
// ---------------------------------------------------------------------------
__global__ void __launch_bounds__(256) z4_kernel(const float* __restrict__ cm,
                                                 const float* __restrict__ bi,
                                                 const float* __restrict__ rmnext,
                                                 const float* __restrict__ cw,
                                                 const float* __restrict__ cb,
                                                 const float* __restrict__ bw,
                                                 const float* __restrict__ bbias,
                                                 const float* __restrict__ u2,
                                                 float* __restrict__ z4out,
                                                 float* __restrict__ outb,
                                                 int O, int Onext, int K) {
    int m = blockIdx.x * 256 + threadIdx.x;
    int b = blockIdx.y;
    if (m >= O) return;
    float a4[16], a1[16];
    for (int o = 0; o < 16; ++o) { a4[o] = cb[o]; a1[o] = bbias[o]; }
    float invOn = (Onext > 0) ? 1.f / (float)Onext : 0.f;
    for (int k = 0; k < K; ++k) {
        float x;
        if (k < 16)      x = cm[(long)(b * 16 + k) * O + m];
        else if (k < 32) x = bi[(long)(b * 16 + k - 16) * O + m];
        else             x = rmnext[(long)(b * 16 + k - 32) * O + m] * invOn;
        for (int o = 0; o < 16; ++o) {
            a4[o] += cw[o * K + k] * x;
            a1[o] += bw[o * K + k] * x;
        }
    }
    for (int o = 0; o < 16; ++o) {
        z4out[(long)(b * 16 + o) * O + m] = a4[o];
        outb[(long)(b * 16 + o) * O + m]  = a1[o] + u2[b * 16 + o];
    }
}

// ---------------------------------------------------------------------------
// K3: fused big pass, templated on <O,N> so all row/channel strides become
// compile-time 24-bit instruction offsets; only 3 base pointers advance per
// 16-wide tile. Lane-half (h) strides folded into the bases once.
//   C <- z3tot tile (+ z4 broadcast), then 4 chained v_wmma_f32_16x16x4_f32
//   with A = z1_w resident in 8 VGPRs; D stored NT straight to out_w.
// ---------------------------------------------------------------------------
template <int O, int N>
__global__ void __launch_bounds__(256)
fused_wmma_kernel(const float* __restrict__ w,       // [B*16, O, N]
                  const float* __restrict__ z1w,     // [16,16] row-major (o,c)
                  const float* __restrict__ z3tot,   // [B,16,N]
                  const float* __restrict__ z4,      // [B,16,O]
                  float* __restrict__ out) {         // [B,16,O,N]
    constexpr long ON = (long)O * N;
    constexpr int NROWS = 8 * O;                     // B * O
    int gw = blockIdx.x * 8 + (threadIdx.x >> 5);
    if (gw >= NROWS) return;                         // wave-uniform
    int b = gw / O, m = gw % O;
    int lane = threadIdx.x & 31;
    int l16 = lane & 15, h = lane >> 4;

    // A matrix (16x4 f32 layout: v0 = K 0/2 by lane half, v1 = K 1/3)
    v2f a0 = *(const v2f*)(z1w + l16 * 16 + 0  + 2 * h);
    v2f a1 = *(const v2f*)(z1w + l16 * 16 + 4  + 2 * h);
    v2f a2 = *(const v2f*)(z1w + l16 * 16 + 8  + 2 * h);
    v2f a3 = *(const v2f*)(z1w + l16 * 16 + 12 + 2 * h);

    float z4v[8];
#pragma unroll
    for (int r = 0; r < 8; ++r)
        z4v[r] = z4[(long)(b * 16 + r + 8 * h) * O + m];

    // fold batch, row m, and lane-half strides into the bases once
    const float* pw  = w     + (long)b * 16 * ON + (long)m * N + (long)(2 * h) * ON + l16;
    const float* pz3 = z3tot + (long)b * 16 * N  + (long)(8 * h) * N + l16;
    float*       po  = out   + (long)b * 16 * ON + (long)m * N + (long)(8 * h) * ON + l16;

    for (int t = 0; t < N / 16; ++t) {
        v8f c;
#pragma unroll
        for (int r = 0; r < 8; ++r)
            c[r] = pz3[(long)r * N] + z4v[r];

        v2f bv0, bv1, bv2, bv3;
        bv0.x = pw[0 * ON];  bv0.y = pw[1 * ON];
        bv1.x = pw[4 * ON];  bv1.y = pw[5 * ON];
        bv2.x = pw[8 * ON];  bv2.y = pw[9 * ON];
        bv3.x = pw[12 * ON]; bv3.y = pw[13 * ON];

        c = __builtin_amdgcn_wmma_f32_16x16x4_f32(false, a0, false, bv0, (short)0, c, false, false);
        c = __builtin_amdgcn_wmma_f32_16x16x4_f32(false, a1, false, bv1, (short)0, c, false, false);
        c = __builtin_amdgcn_wmma_f32_16x16x4_f32(false, a2, false, bv2, (short)0, c, false, false);
        c = __builtin_amdgcn_wmma_f32_16x16x4_f32(false, a3, false, bv3, (short)0, c, false, false);

#pragma unroll
        for (int r = 0; r < 8; ++r)
            __builtin_nontemporal_store(c[r], po + (long)r * ON);   // TH=NT: never re-read

        pw += 16; pz3 += 16; po += 16;
    }
}

// ---------------------------------------------------------------------------
// kernel_launch
// ---------------------------------------------------------------------------
extern "C" void kernel_launch(void* const* d_in, const int* in_sizes, int n_in,
                              void* d_out, int out_size, void* d_ws, size_t ws_size,
                              hipStream_t stream) {
    const float* w[3]  = {(const float*)d_in[0], (const float*)d_in[2], (const float*)d_in[4]};
    const float* bi[3] = {(const float*)d_in[1], (const float*)d_in[3], (const float*)d_in[5]};
    const float* prm[36];
    for (int i = 0; i < 36; ++i) prm[i] = (const float*)d_in[6 + i];
    // per-layer param slots: 0:z1_w 1:z1_b 2:rc_w 3:rc_b 4:r_w 5:r_b 6:c_w 7:c_b
    //                        8:bias_w 9:bias_b 10:bias_rc_w 11:bias_rc_b
    #define P(i, j) (prm[12 * (i) + (j)])

    float* ws = (float*)d_ws;
    // workspace layout (float offsets)
    const long off_rm[3] = {0, 100352, 165888};          // row-sum accumulators
    const long off_rc    = 231424;                       // rowcol-sum accumulators (3*128)
    const long n_accum   = 231808;                       // region to zero
    const long off_cm[3] = {231808, 297344, 362880};     // col means
    const long off_wb    = 364160;                       // wb_means [8][96]
    const long off_z2    = 364928;                       // z2 [3][8][16]
    const long off_u2    = 365312;                       // u2 [3][8][16]
    const long off_z3[3] = {365696, 466048, 531584};     // z3tot
    const long off_z4[3] = {597120, 662656, 728192};     // z4

    float* out = (float*)d_out;
    const long off_ow[3] = {0, 51380224, 84934656};
    const long off_ob[3] = {85590016, 85655552, 85721088};

    // pass 0: zero accumulators
    zero_kernel<<<(unsigned)((n_accum + 255) / 256), 256, 0, stream>>>(ws, n_accum);

    // pass 1: reductions (one read of each weight tensor)
    reduce_kernel<512, 784><<<dim3(8, 128), 256, 0, stream>>>(w[0], ws + off_rm[0], ws + off_cm[0], ws + off_rc + 0);
    reduce_kernel<512, 512><<<dim3(8, 128), 256, 0, stream>>>(w[1], ws + off_rm[1], ws + off_cm[1], ws + off_rc + 128);
    reduce_kernel<10,  512><<<dim3(1, 128), 256, 0, stream>>>(w[2], ws + off_rm[2], ws + off_cm[2], ws + off_rc + 256);

    // pass 2: small derived quantities
    wb_kernel<<<3, 256, 0, stream>>>(ws + off_rc, bi[0], bi[1], bi[2], ws + off_wb);

    z2u2_kernel<<<2, 256, 0, stream>>>(ws + off_wb,
        P(0, 2), P(0, 3), P(1, 2), P(1, 3), P(2, 2), P(2, 3),
        P(0, 10), P(0, 11), P(1, 10), P(1, 11), P(2, 10), P(2, 11),
        ws + off_z2, ws + off_u2);

    // z3tot: row broadcast terms (+ z1_b + z2 folded in)
    z3_kernel<<<dim3(4, 8), 256, 0, stream>>>(ws + off_rm[0], nullptr, nullptr,
        P(0, 4), P(0, 5), P(0, 1), ws + off_z2 + 0,   ws + off_z3[0], 784, 512, 0,   16);
    z3_kernel<<<dim3(2, 8), 256, 0, stream>>>(ws + off_rm[1], ws + off_cm[0], bi[0],
        P(1, 4), P(1, 5), P(1, 1), ws + off_z2 + 128, ws + off_z3[1], 512, 512, 512, 48);
    z3_kernel<<<dim3(2, 8), 256, 0, stream>>>(ws + off_rm[2], ws + off_cm[1], bi[1],
        P(2, 4), P(2, 5), P(2, 1), ws + off_z2 + 256, ws + off_z3[2], 512, 10,  512, 48);

    // z4 (col broadcast terms) + out_b (u1 + u2) in one pass
    z4_kernel<<<dim3(2, 8), 256, 0, stream>>>(ws + off_cm[0], bi[0], ws + off_rm[1],
        P(0, 6), P(0, 7), P(0, 8), P(0, 9), ws + off_u2 + 0,   ws + off_z4[0], out + off_ob[0], 512, 512, 48);
    z4_kernel<<<dim3(2, 8), 256, 0, stream>>>(ws + off_cm[1], bi[1], ws + off_rm[2],
        P(1, 6), P(1, 7), P(1, 8), P(1, 9), ws + off_u2 + 128, ws + off_z4[1], out + off_ob[1], 512, 10,  48);
    z4_kernel<<<dim3(1, 8), 256, 0, stream>>>(ws + off_cm[2], bi[2], nullptr,
        P(2, 6), P(2, 7), P(2, 8), P(2, 9), ws + off_u2 + 256, ws + off_z4[2], out + off_ob[2], 10,  0,   32);

    // pass 3: fused z1 WMMA + broadcast adds -> out_w (BW-bound main pass)
    fused_wmma_kernel<512, 784><<<512, 256, 0, stream>>>(w[0], P(0, 0), ws + off_z3[0], ws + off_z4[0], out + off_ow[0]);
    fused_wmma_kernel<512, 512><<<512, 256, 0, stream>>>(w[1], P(1, 0), ws + off_z3[1], ws + off_z4[1], out + off_ow[1]);
    fused_wmma_kernel<10,  512><<<10,  256, 0, stream>>>(w[2], P(2, 0), ws + off_z3[2], ws + off_z4[2], out + off_ow[2]);
    #undef P
}